// RPN_Loss_50869592654320
// MI455X (gfx1250) — compile-verified
//
#include <hip/hip_runtime.h>

#define NBINS       4096
#define P1_THREADS  1024
#define P1_ITERS    16
#define ANCH_PER_BLOCK (P1_THREADS * 2 * P1_ITERS)   // 32768 anchors / block

typedef __attribute__((ext_vector_type(2))) float v2f;
typedef __attribute__((ext_vector_type(8))) float v8f;

// Exact f32 sum of all 64 accumulators (2 per lane) across a wave32 using the
// matrix pipe: D[m,n] = sum_k A[m,k] * 1.0, then 8 adds + one xor-16 shuffle.
__device__ __forceinline__ float wave_sum_pair(float a0, float a1) {
    v2f a; a[0] = a0;   a[1] = a1;
    v2f b; b[0] = 1.0f; b[1] = 1.0f;
    v8f c = {};
    v8f d = __builtin_amdgcn_wmma_f32_16x16x4_f32(
        /*neg_a=*/false, a, /*neg_b=*/false, b,
        /*c_mod=*/(short)0, c, /*reuse_a=*/false, /*reuse_b=*/false);
    float s = d[0] + d[1] + d[2] + d[3] + d[4] + d[5] + d[6] + d[7];
    s += __shfl_xor(s, 16, 32);
    return s;
}

__global__ void zero_ws_kernel(unsigned* w, int n) {
    int i = blockIdx.x * blockDim.x + threadIdx.x;
    if (i < n) w[i] = 0u;
}

__global__ __launch_bounds__(P1_THREADS)
void rpn_pass1(const float* __restrict__ cls,          // [N,2]
               const long long* __restrict__ tgt,      // [N]
               long long nAnchors,
               unsigned* __restrict__ gCnt,            // [NBINS]
               float* __restrict__ gSum,               // [NBINS]
               float* __restrict__ gPosStats) {        // [2]: posSum, posCnt
    __shared__ unsigned hC[NBINS];
    __shared__ float    hS[NBINS];
    const int tid = threadIdx.x;

    for (int i = tid; i < NBINS; i += P1_THREADS) { hC[i] = 0u; hS[i] = 0.0f; }
    __syncthreads();

    float accPos0 = 0.0f, accPos1 = 0.0f;   // alternating accumulators -> WMMA A
    float cntPos0 = 0.0f, cntPos1 = 0.0f;

    const long long base = (long long)blockIdx.x * ANCH_PER_BLOCK;

    for (int it = 0; it < P1_ITERS; ++it) {
        long long a = base + (long long)it * (P1_THREADS * 2) + (long long)tid * 2;
        if (a + 1 < nAnchors) {
            // prefetch next iteration's cachelines (global_prefetch_b8)
            __builtin_prefetch(cls + (a + P1_THREADS * 2) * 2, 0, 1);
            __builtin_prefetch(tgt + (a + P1_THREADS * 2), 0, 1);

            float4    l  = *reinterpret_cast<const float4*>(cls + a * 2);
            longlong2 lb = *reinterpret_cast<const longlong2*>(tgt + a);

            // anchor a
            {
                float x = l.x, y = l.y;
                float m = fmaxf(x, y);
                float lse = m + __logf(__expf(x - m) + __expf(y - m));
                int lab = (int)lb.x;
                if (lab == 1) { accPos0 += (lse - y); cntPos0 += 1.0f; }
                else if (lab == 0) {
                    float cen = lse - x;                      // >= 0
                    unsigned bin = __float_as_uint(cen) >> 19; // exp + 4 mantissa bits
                    if (bin > NBINS - 1) bin = NBINS - 1;
                    atomicAdd(&hC[bin], 1u);
                    atomicAdd(&hS[bin], cen);
                }
            }
            // anchor a+1
            {
                float x = l.z, y = l.w;
                float m = fmaxf(x, y);
                float lse = m + __logf(__expf(x - m) + __expf(y - m));
                int lab = (int)lb.y;
                if (lab == 1) { accPos1 += (lse - y); cntPos1 += 1.0f; }
                else if (lab == 0) {
                    float cen = lse - x;
                    unsigned bin = __float_as_uint(cen) >> 19;
                    if (bin > NBINS - 1) bin = NBINS - 1;
                    atomicAdd(&hC[bin], 1u);
                    atomicAdd(&hS[bin], cen);
                }
            }
        } else if (a < nAnchors) {               // scalar tail (N odd)
            float2 l2 = *reinterpret_cast<const float2*>(cls + a * 2);
            long long lab = tgt[a];
            float x = l2.x, y = l2.y;
            float m = fmaxf(x, y);
            float lse = m + __logf(__expf(x - m) + __expf(y - m));
            if (lab == 1) { accPos0 += (lse - y); cntPos0 += 1.0f; }
            else if (lab == 0) {
                float cen = lse - x;
                unsigned bin = __float_as_uint(cen) >> 19;
                if (bin > NBINS - 1) bin = NBINS - 1;
                atomicAdd(&hC[bin], 1u);
                atomicAdd(&hS[bin], cen);
            }
        }
    }
    __syncthreads();   // LDS histogram complete; EXEC all-ones from here

    // wave32 reduction on the matrix pipe (2x v_wmma_f32_16x16x4_f32)
    float wPosSum = wave_sum_pair(accPos0, accPos1);
    float wPosCnt = wave_sum_pair(cntPos0, cntPos1);
    if ((tid & 31) == 0) {
        atomicAdd(&gPosStats[0], wPosSum);
        atomicAdd(&gPosStats[1], wPosCnt);   // integer-valued, exact (< 2^24)
    }

    // flush LDS histogram (only non-empty bins)
    for (int i = tid; i < NBINS; i += P1_THREADS) {
        unsigned c = hC[i];
        if (c) {
            atomicAdd(&gCnt[i], c);
            atomicAdd(&gSum[i], hS[i]);
        }
    }
}

__global__ __launch_bounds__(1024)
void rpn_finalize(const unsigned* __restrict__ gCnt,
                  const float* __restrict__ gSum,
                  const float* __restrict__ gPosStats,
                  float* __restrict__ out) {
    __shared__ unsigned sC[NBINS];
    __shared__ float    sS[NBINS];
    __shared__ unsigned sNNeg;
    __shared__ float    sNegAll;
    __shared__ int      sT;
    const int tid = threadIdx.x;

    if (tid == 0) { sNNeg = 0u; sNegAll = 0.0f; sT = -1; }
    unsigned pc = 0; float ps = 0.0f;
    for (int j = 0; j < 4; ++j) {
        int i = tid + j * 1024;
        unsigned c = gCnt[i]; float s = gSum[i];
        sC[i] = c; sS[i] = s;
        pc += c;  ps += s;
    }
    __syncthreads();
    atomicAdd(&sNNeg, pc);
    atomicAdd(&sNegAll, ps);
    __syncthreads();

    const long long nNeg   = (long long)sNNeg;
    const float     negAll = sNegAll;
    const float     posSum = gPosStats[0];
    const long long nPos   = (long long)(gPosStats[1] + 0.5f);
    long long k = 3 * nPos; if (nNeg < k) k = nNeg;

    float negLoss;
    if (k >= nNeg) {
        // common case: all negatives selected -> exact
        negLoss = negAll / (float)k;               // k==0 -> 0/0 = NaN (matches ref)
    } else {
        // rare: in-place Hillis-Steele inclusive suffix scans over 4096 bins
        for (int off = 1; off < NBINS; off <<= 1) {
            unsigned tc[4]; float ts[4];
            for (int j = 0; j < 4; ++j) {
                int i = tid + j * 1024;
                bool ok = (i + off) < NBINS;
                tc[j] = ok ? sC[i + off] : 0u;
                ts[j] = ok ? sS[i + off] : 0.0f;
            }
            __syncthreads();
            for (int j = 0; j < 4; ++j) {
                int i = tid + j * 1024;
                sC[i] += tc[j];
                sS[i] += ts[j];
            }
            __syncthreads();
        }
        // threshold bin t: suffixCnt[t] >= k > suffixCnt[t+1]
        for (int j = 0; j < 4; ++j) {
            int i = tid + j * 1024;
            long long Si = (long long)sC[i];
            long long Sn = (i + 1 < NBINS) ? (long long)sC[i + 1] : 0;
            if (Si >= k && Sn < k) sT = i;   // unique writer
        }
        __syncthreads();
        int t = sT;
        if (t < 0) {
            negLoss = 0.0f / 0.0f;           // k<=0 -> NaN like reference
        } else {
            long long above = (t + 1 < NBINS) ? (long long)sC[t + 1] : 0;
            float sumAbove  = (t + 1 < NBINS) ? sS[t + 1] : 0.0f;
            long long r = k - above;         // 1 <= r <= gCnt[t]
            float binMean = gSum[t] / (float)gCnt[t];
            negLoss = (sumAbove + (float)r * binMean) / (float)k;
        }
    }

    if (tid == 0) out[0] = posSum / (float)nPos + negLoss;
}

extern "C" void kernel_launch(void* const* d_in, const int* in_sizes, int n_in,
                              void* d_out, int out_size, void* d_ws, size_t ws_size,
                              hipStream_t stream) {
    const float*     cls = (const float*)d_in[0];        // [1,N,2] f32
    const long long* tgt = (const long long*)d_in[3];    // [1,1,N] i64
    const long long  N   = (long long)in_sizes[3];

    unsigned* gCnt     = (unsigned*)d_ws;
    float*    gSum     = (float*)((char*)d_ws + NBINS * sizeof(unsigned));
    float*    gPosStat = (float*)((char*)d_ws + 2 * NBINS * sizeof(unsigned));

    const int wsWords = 2 * NBINS + 2;
    zero_ws_kernel<<<(wsWords + 255) / 256, 256, 0, stream>>>((unsigned*)d_ws, wsWords);

    const int blocks = (int)((N + ANCH_PER_BLOCK - 1) / ANCH_PER_BLOCK);
    rpn_pass1<<<blocks, P1_THREADS, 0, stream>>>(cls, tgt, N, gCnt, gSum, gPosStat);

    rpn_finalize<<<1, 1024, 0, stream>>>(gCnt, gSum, gPosStat, (float*)d_out);
}